// TopKRank_17703855194721
// MI455X (gfx1250) — compile-verified
//
#include <hip/hip_runtime.h>

typedef __attribute__((ext_vector_type(2))) float v2f;
typedef __attribute__((ext_vector_type(4))) float v4f;
typedef __attribute__((ext_vector_type(8))) float v8f;

// ---------------------------------------------------------------------------
// Pass 1: spatial mean per (b,c) via V_WMMA_F32_16X16X4_F32 with B = ones.
// One wave per channel plane (4096 floats). Each lane loads a float4
// (coalesced 512B per wave-load); the 128 floats feed two WMMAs whose D
// accumulates row-sums (B is all ones, so the lane->element mapping is
// irrelevant for the total). Total = sum of 8 acc VGPRs + shfl_xor(16).
// ---------------------------------------------------------------------------
__global__ void eca_mean_kernel(const float* __restrict__ x,
                                float* __restrict__ y) {
    const int wave = threadIdx.x >> 5;
    const int lane = threadIdx.x & 31;
    const int bc   = blockIdx.x * 8 + wave;          // 0 .. 16383  (b*512+c)

    const v4f* __restrict__ p4 = (const v4f*)(x + (size_t)bc * 4096);

    v8f acc = {};
    v2f ones;
    ones[0] = 1.0f; ones[1] = 1.0f;

#pragma unroll 4
    for (int g = 0; g < 32; ++g) {
        v4f v = p4[g * 32 + lane];                   // 128 floats / wave / iter
        v2f a0; a0[0] = v.x; a0[1] = v.y;
        v2f a1; a1[0] = v.z; a1[1] = v.w;
        acc = __builtin_amdgcn_wmma_f32_16x16x4_f32(
            false, a0, false, ones, (short)0, acc, false, false);
        acc = __builtin_amdgcn_wmma_f32_16x16x4_f32(
            false, a1, false, ones, (short)0, acc, false, false);
    }

    // acc[i] on lanes 0-15 holds D[i][n] (rows 0-7), lanes 16-31 rows 8-15;
    // all columns identical. Total = low-half rows + high-half rows.
    float s = acc[0] + acc[1] + acc[2] + acc[3] +
              acc[4] + acc[5] + acc[6] + acc[7];
    s += __shfl_xor(s, 16, 32);                      // wave32 cross-half add

    if (lane == 0) y[bc] = s * (1.0f / 4096.0f);
}

// ---------------------------------------------------------------------------
// Pass 2: conv1d(k=3, pad=1) over channels + sigmoid, then exact top-k rank.
// One block per batch, 512 threads (one per channel). Rank of channel c =
// #{j : s_j > s_c or (s_j == s_c and j < c)} == its position in the
// descending stable sort, matching jax.lax.top_k tie-breaking.
// ---------------------------------------------------------------------------
__global__ void eca_topk_kernel(const float* __restrict__ y,
                                const float* __restrict__ w,
                                int* __restrict__ idx) {
    __shared__ float sh[512];
    const int b = blockIdx.x;
    const int c = threadIdx.x;
    const float* __restrict__ yb = y + b * 512;

    const float w0 = w[0], w1 = w[1], w2 = w[2];
    const float ym1 = (c > 0)   ? yb[c - 1] : 0.0f;  // zero padding
    const float y0  = yb[c];
    const float yp1 = (c < 511) ? yb[c + 1] : 0.0f;

    const float logit = w0 * ym1 + w1 * y0 + w2 * yp1;
    const float s = 1.0f / (1.0f + __expf(-logit));

    sh[c] = s;
    __syncthreads();

    int rank = 0;
    for (int j = 0; j < 512; ++j) {
        const float sj = sh[j];
        rank += (sj > s) || (sj == s && j < c);
    }
    if (rank < 256) idx[b * 256 + rank] = c;
}

// ---------------------------------------------------------------------------
// Pass 3: batched channel gather. Each block copies one 16 KB channel plane
// (b, idx[b][j]) -> out(b, j) with 128-bit loads and nontemporal 128-bit
// stores (output is write-once, keep it out of cache).
// ---------------------------------------------------------------------------
__global__ void eca_gather_kernel(const float* __restrict__ x,
                                  const int* __restrict__ idx,
                                  float* __restrict__ out) {
    const int b = blockIdx.z;
    const int j = blockIdx.y;
    const int c = idx[b * 256 + j];

    const v4f* __restrict__ src =
        (const v4f*)(x + ((size_t)(b * 512 + c)) * 4096);
    v4f* __restrict__ dst =
        (v4f*)(out + ((size_t)(b * 256 + j)) * 4096);

    const int t = threadIdx.x;
#pragma unroll
    for (int i = 0; i < 4; ++i) {
        v4f v = src[i * 256 + t];
        __builtin_nontemporal_store(v, &dst[i * 256 + t]);
    }
}

// ---------------------------------------------------------------------------
// Launch: mean (16384 waves) -> scores/top-k (32 blocks) -> gather (8192
// blocks). Workspace: y = 32*512 floats (64 KB), then idx = 32*256 ints.
// ---------------------------------------------------------------------------
extern "C" void kernel_launch(void* const* d_in, const int* in_sizes, int n_in,
                              void* d_out, int out_size, void* d_ws, size_t ws_size,
                              hipStream_t stream) {
    const float* x = (const float*)d_in[0];   // [32, 512, 64, 64] fp32
    const float* w = (const float*)d_in[1];   // [3] fp32
    float* out = (float*)d_out;               // [32, 256, 64, 64] fp32

    float* y   = (float*)d_ws;                                   // 16384 floats
    int*   idx = (int*)((char*)d_ws + 32 * 512 * sizeof(float)); // 8192 ints

    eca_mean_kernel<<<2048, 256, 0, stream>>>(x, y);
    eca_topk_kernel<<<32, 512, 0, stream>>>(y, w, idx);
    eca_gather_kernel<<<dim3(1, 256, 32), 256, 0, stream>>>(x, idx, out);
}